// DecodeSBP_6708738916374
// MI455X (gfx1250) — compile-verified
//
#include <hip/hip_runtime.h>
#include <hip/hip_bf16.h>

// Problem constants (from reference):
//   x: [1, 133, 512, 512] fp32 ; out: [133, 3] fp32
//   sigmoid is monotone -> argmax on raw logits, sigmoid only on final max.
#define KPTS        133
#define WIMG        512
#define HW          (512 * 512)        // 262144
#define CHUNKS      32                 // blocks per heatmap
#define CHUNK_ELEMS (HW / CHUNKS)      // 8192 floats per block
#define TILES       8                  // tiles per block (each tile = 1024 floats = 4KB)
#define CONF_THRESH 0.8f
#define SCALE       4.0f               // 2048 / 512

// ---------------------------------------------------------------------------
// Kernel 1: per-chunk argmax partials, streamed through LDS with the CDNA5
// async path (GLOBAL_LOAD_ASYNC_TO_LDS_B128 + s_wait_asynccnt).
// Fully-buffered: each wave owns 8 private 512B LDS landing slots, issues all
// 8 async B128 loads up front (max MLP, no LDS slot reuse -> no WAR window),
// then drains in order with progressively smaller ASYNCcnt waits.
// No block barriers needed: each wave touches only its own LDS slice and its
// own ASYNCcnt.
// ---------------------------------------------------------------------------
__global__ __launch_bounds__(256) void
DecodeSBP_argmax_part(const float* __restrict__ x,
                      float* __restrict__ pval,
                      int* __restrict__ pidx) {
    // 8 waves * 8 tiles * 128 floats = 8192 floats = 32KB landing buffers
    __shared__ __align__(16) float lds[8192];
    __shared__ float s_val[8];
    __shared__ int   s_idx[8];

    const int bid = blockIdx.x;
    const int k   = bid >> 5;          // heatmap index
    const int c   = bid & 31;          // chunk within heatmap
    const int w   = threadIdx.x >> 5;  // wave in block (wave32)
    const int l   = threadIdx.x & 31;  // lane

    const int lane_off = (w * 32 + l) * 4;   // float offset of this lane within a tile
    const float* gp = x + (size_t)k * HW + (size_t)c * CHUNK_ELEMS + lane_off;

    // LDS byte offset of this lane's tile-0 slot
    // (low 32 bits of a generic LDS pointer == AS(3) byte offset).
    const unsigned lds0 = (unsigned)(size_t)&lds[w * 1024 + l * 4];

    // Prologue: issue ALL 8 tiles (8 outstanding async B128 loads per lane).
    {
        unsigned long long ga = (unsigned long long)(const void*)gp;
        unsigned dst = lds0;
        #pragma unroll
        for (int t = 0; t < TILES; ++t) {
            asm volatile("global_load_async_to_lds_b128 %0, %1, off"
                         :: "v"(dst), "v"(ga) : "memory");
            ga  += 4096ull;   // 1024 floats per tile
            dst += 512u;      // 128 floats per wave-slot
        }
    }

    float best = -__builtin_inff();
    int   bidx = 0;

    #pragma unroll
    for (int j = 0; j < TILES; ++j) {
        // ASYNCcnt completes loads in order -> cnt <= (7-j) means tile j done.
        if      (j == 0) asm volatile("s_wait_asynccnt 0x7" ::: "memory");
        else if (j == 1) asm volatile("s_wait_asynccnt 0x6" ::: "memory");
        else if (j == 2) asm volatile("s_wait_asynccnt 0x5" ::: "memory");
        else if (j == 3) asm volatile("s_wait_asynccnt 0x4" ::: "memory");
        else if (j == 4) asm volatile("s_wait_asynccnt 0x3" ::: "memory");
        else if (j == 5) asm volatile("s_wait_asynccnt 0x2" ::: "memory");
        else if (j == 6) asm volatile("s_wait_asynccnt 0x1" ::: "memory");
        else             asm volatile("s_wait_asynccnt 0x0" ::: "memory");

        // Read this lane's 16B of tile j from LDS (ds_load_b128).
        const float4 v = *(const float4*)&lds[w * 1024 + j * 128 + l * 4];

        // Global flat index (within heatmap) of v.x; per-thread indices are
        // strictly increasing across j and x..w, so strict '>' keeps the
        // first occurrence (matches jnp.argmax tie-breaking).
        const int gi = c * CHUNK_ELEMS + j * 1024 + lane_off;
        if (v.x > best) { best = v.x; bidx = gi; }
        if (v.y > best) { best = v.y; bidx = gi + 1; }
        if (v.z > best) { best = v.z; bidx = gi + 2; }
        if (v.w > best) { best = v.w; bidx = gi + 3; }
    }

    // Wave32 reduction, tiebreak on smaller index for equal values.
    #pragma unroll
    for (int off = 16; off > 0; off >>= 1) {
        const float ov = __shfl_down(best, off, 32);
        const int   oi = __shfl_down(bidx, off, 32);
        if (ov > best || (ov == best && oi < bidx)) { best = ov; bidx = oi; }
    }
    if (l == 0) { s_val[w] = best; s_idx[w] = bidx; }
    __syncthreads();

    if (threadIdx.x == 0) {
        float bv = s_val[0];
        int   bi = s_idx[0];
        #pragma unroll
        for (int i = 1; i < 8; ++i) {
            if (s_val[i] > bv || (s_val[i] == bv && s_idx[i] < bi)) {
                bv = s_val[i]; bi = s_idx[i];
            }
        }
        pval[bid] = bv;
        pidx[bid] = bi;
    }
}

// ---------------------------------------------------------------------------
// Kernel 2: one wave per keypoint reduces its 32 chunk-partials, applies the
// sigmoid once, thresholds, and writes the [x*4, y*4, conf] row.
// ---------------------------------------------------------------------------
__global__ __launch_bounds__(32) void
DecodeSBP_finalize(const float* __restrict__ pval,
                   const int* __restrict__ pidx,
                   float* __restrict__ out) {
    const int k = blockIdx.x;
    const int l = threadIdx.x;

    float v = pval[k * CHUNKS + l];
    int   i = pidx[k * CHUNKS + l];

    #pragma unroll
    for (int off = 16; off > 0; off >>= 1) {
        const float ov = __shfl_down(v, off, 32);
        const int   oi = __shfl_down(i, off, 32);
        if (ov > v || (ov == v && oi < i)) { v = ov; i = oi; }
    }

    if (l == 0) {
        const float conf = 1.0f / (1.0f + expf(-v));   // sigmoid(max logit)
        const float xx = (float)(i & (WIMG - 1));
        const float yy = (float)(i >> 9);
        const bool valid = conf > CONF_THRESH;
        out[k * 3 + 0] = valid ? xx * SCALE : -SCALE;  // (-1)*scale for invalid
        out[k * 3 + 1] = valid ? yy * SCALE : -SCALE;
        out[k * 3 + 2] = valid ? conf : -1.0f;
    }
}

// ---------------------------------------------------------------------------
extern "C" void kernel_launch(void* const* d_in, const int* in_sizes, int n_in,
                              void* d_out, int out_size, void* d_ws, size_t ws_size,
                              hipStream_t stream) {
    (void)in_sizes; (void)n_in; (void)out_size; (void)ws_size;

    const float* x = (const float*)d_in[0];
    float* out = (float*)d_out;

    const int nparts = KPTS * CHUNKS;                   // 4256
    float* pval = (float*)d_ws;
    int*   pidx = (int*)((char*)d_ws + (size_t)nparts * sizeof(float));

    DecodeSBP_argmax_part<<<dim3(nparts), dim3(256), 0, stream>>>(x, pval, pidx);
    DecodeSBP_finalize<<<dim3(KPTS), dim3(32), 0, stream>>>(pval, pidx, out);
}